// RecyclingEmbedder_73796128080194
// MI455X (gfx1250) — compile-verified
//
#include <hip/hip_runtime.h>
#include <stdint.h>

// ---------------------------------------------------------------------------
// RecyclingEmbedder for MI455X (gfx1250, wave32).
// Memory-bound: ~605 MB streamed -> ~26 us at 23.3 TB/s. The one-hot
// distogram einsum is a row-gather of lin_w^T (no real GEMM), so the CDNA5
// feature that pays off is the async global->LDS copy engine:
//   global_load_async_to_lds_b128 (NT hint) + s_wait_asynccnt,
// run as a 4-deep per-wave ring buffer, plus non-temporal b128 stores and
// ds_swizzle-based wave32 butterfly reductions for the LayerNorms.
// ---------------------------------------------------------------------------

#define CM     256
#define CZ     128
#define NB     15
#define NN     768
#define NROWS  (NN * NN)           // 589824 pair rows
#define OUT_Z_OFF (NN * CM)        // 196608 floats of m_update first
#define LN_EPS 1e-5f

typedef float f4 __attribute__((ext_vector_type(4)));

// xor-butterfly add via ds_swizzle_b32 (group-of-32: and=0x1f, or=0, xor=MASK)
template <int MASK>
__device__ __forceinline__ float swz_add(float v) {
  constexpr int off = (MASK << 10) | 0x1f;
  const int t = __builtin_amdgcn_ds_swizzle(__float_as_int(v), off);
  return v + __int_as_float(t);
}

__device__ __forceinline__ float wave_sum32(float v) {
  v = swz_add<16>(v);
  v = swz_add<8>(v);
  v = swz_add<4>(v);
  v = swz_add<2>(v);
  v = swz_add<1>(v);
  return v;
}

// Low 32 bits of a generic pointer to LDS == LDS byte offset (flat-LDS rule).
__device__ __forceinline__ uint32_t lds_off32(const void* p) {
  return (uint32_t)(uintptr_t)p;
}

// -------------------------- m LayerNorm kernel -----------------------------
__global__ __launch_bounds__(256) void m_ln_kernel(
    const float* __restrict__ m, const float* __restrict__ w,
    const float* __restrict__ b, float* __restrict__ out) {
  __shared__ float red[16];
  const int row  = blockIdx.x;
  const int tid  = threadIdx.x;
  const int lane = tid & 31;
  const int wid  = tid >> 5;

  const float v = m[row * CM + tid];
  float s  = wave_sum32(v);
  float s2 = wave_sum32(v * v);
  if (lane == 0) { red[wid] = s; red[8 + wid] = s2; }
  __syncthreads();
  if (wid == 0) {
    float a  = (lane < 8) ? red[lane]     : 0.0f;
    float a2 = (lane < 8) ? red[8 + lane] : 0.0f;
    a  = swz_add<4>(a);  a  = swz_add<2>(a);  a  = swz_add<1>(a);
    a2 = swz_add<4>(a2); a2 = swz_add<2>(a2); a2 = swz_add<1>(a2);
    if (lane == 0) { red[0] = a; red[1] = a2; }
  }
  __syncthreads();
  const float mean = red[0] * (1.0f / 256.0f);
  const float var  = red[1] * (1.0f / 256.0f) - mean * mean;
  const float inv  = __frsqrt_rn(var + LN_EPS);
  out[row * CM + tid] = (v - mean) * inv * w[tid] + b[tid];
}

// ---------------- z: distogram-embed + LayerNorm, async-LDS pipeline -------
__global__ __launch_bounds__(256) void z_kernel(
    const float* __restrict__ z, const float* __restrict__ x,
    const float* __restrict__ lin_w, const float* __restrict__ lin_b,
    const float* __restrict__ lzw, const float* __restrict__ lzb,
    float* __restrict__ out) {
  __shared__ __align__(16) float xl[NN * 3];        // 9216 B
  __shared__ __align__(16) float lwT[NB * CZ];      // lin_w transposed, 7680 B
  __shared__ __align__(16) float lzw_s[CZ];
  __shared__ __align__(16) float lzb_s[CZ];
  __shared__ __align__(16) float lb_s[CZ];
  __shared__ __align__(16) float abuf[8 * 4 * CZ];  // 8 waves x 4 slots x 512 B

  const int tid = threadIdx.x;
  for (int idx = tid; idx < NN * 3; idx += 256) xl[idx] = x[idx];
  for (int idx = tid; idx < CZ * NB; idx += 256) {
    const int c = idx / NB, b = idx - c * NB;
    lwT[b * CZ + c] = lin_w[idx];
  }
  if (tid < CZ) { lzw_s[tid] = lzw[tid]; lzb_s[tid] = lzb[tid]; lb_s[tid] = lin_b[tid]; }
  __syncthreads();

  const int lane = tid & 31;
  const int wid  = tid >> 5;
  const int gw   = blockIdx.x * 8 + wid;   // global wave id
  const int nW   = gridDim.x * 8;          // total waves

  const uint64_t zbase   = (uint64_t)(uintptr_t)z;
  // per-lane LDS destination base for this wave's 4-slot ring
  const uint32_t ldsbase = lds_off32(&abuf[0]) + (uint32_t)(wid * 2048 + lane * 16);

  // squared bin edges: linspace(3.25, 20.75, 15)^2, upper[14] = 1e8
  const float SQB[16] = {10.5625f, 20.25f,  33.0625f, 49.0f,   68.0625f,
                         90.25f,   115.5625f, 144.0f, 175.5625f, 210.25f,
                         248.0625f, 289.0f, 333.0625f, 380.25f, 430.5625f, 1e8f};

  auto issue = [&](int row, int slot) {
    const uint32_t voff = (uint32_t)row * 512u + (uint32_t)(lane * 16);
    const uint32_t dst  = ldsbase + (uint32_t)(slot * 512);
    asm volatile("global_load_async_to_lds_b128 %0, %1, %2 th:TH_LOAD_NT"
                 :: "v"(dst), "v"(voff), "s"(zbase) : "memory");
  };

  if (gw >= NROWS) return;

  // prologue: 3 rows in flight (slots 0..2)
#pragma unroll
  for (int p = 0; p < 3; ++p) {
    int rr = gw + p * nW;
    if (rr >= NROWS) rr = gw;   // clamp keeps the wait-count invariant
    issue(rr, p);
  }

  int k = 0;
  for (int r = gw; r < NROWS; r += nW, ++k) {
    // 3 loads outstanding here; oldest is row r
    asm volatile("s_wait_asynccnt 2" ::: "memory");
    const f4 v = *(const f4*)&abuf[wid * 512 + (k & 3) * 128 + lane * 4];
    // make sure the LDS read retired before this slot is re-targeted
    asm volatile("s_wait_dscnt 0" ::: "memory");
    int rn = r + 3 * nW;
    if (rn >= NROWS) rn = r;
    issue(rn, (k + 3) & 3);     // back to 3 outstanding

    // ---- distogram bin (wave-uniform) ----
    const int i = r / NN;
    const int j = r - i * NN;
    const float dx = xl[i * 3 + 0] - xl[j * 3 + 0];
    const float dy = xl[i * 3 + 1] - xl[j * 3 + 1];
    const float dz = xl[i * 3 + 2] - xl[j * 3 + 2];
    const float d  = dx * dx + dy * dy + dz * dz;
    int bin = -1;
#pragma unroll
    for (int b = 0; b < 15; ++b)
      if (d > SQB[b] && d < SQB[b + 1]) bin = b;

    // ---- LayerNorm over 128 channels (32 lanes x 4) ----
    const float s  = wave_sum32(v.x + v.y + v.z + v.w);
    const float s2 = wave_sum32(v.x * v.x + v.y * v.y + v.z * v.z + v.w * v.w);
    const float mean = s * (1.0f / 128.0f);
    const float var  = s2 * (1.0f / 128.0f) - mean * mean;
    const float inv  = __frsqrt_rn(var + LN_EPS);

    const int c0 = lane * 4;
    f4 e;
    if (bin >= 0) e = *(const f4*)&lwT[bin * CZ + c0];
    else          e = (f4){0.0f, 0.0f, 0.0f, 0.0f};
    const f4 w  = *(const f4*)&lzw_s[c0];
    const f4 bb = *(const f4*)&lzb_s[c0];
    const f4 lb = *(const f4*)&lb_s[c0];

    f4 o;
    o.x = (v.x - mean) * inv * w.x + bb.x + e.x + lb.x;
    o.y = (v.y - mean) * inv * w.y + bb.y + e.y + lb.y;
    o.z = (v.z - mean) * inv * w.z + bb.z + e.z + lb.z;
    o.w = (v.w - mean) * inv * w.w + bb.w + e.w + lb.w;

    // streaming output: non-temporal b128 store
    __builtin_nontemporal_store(
        o, (f4*)&out[(size_t)OUT_Z_OFF + (size_t)r * CZ + c0]);
  }
}

// ---------------------------------------------------------------------------
extern "C" void kernel_launch(void* const* d_in, const int* in_sizes, int n_in,
                              void* d_out, int out_size, void* d_ws, size_t ws_size,
                              hipStream_t stream) {
  (void)in_sizes; (void)n_in; (void)out_size; (void)d_ws; (void)ws_size;
  const float* m      = (const float*)d_in[0];
  const float* z      = (const float*)d_in[1];
  const float* x      = (const float*)d_in[2];
  /* d_in[3] = seqs, unused by reference */
  const float* lin_w  = (const float*)d_in[4];
  const float* lin_b  = (const float*)d_in[5];
  const float* ln_m_w = (const float*)d_in[6];
  const float* ln_m_b = (const float*)d_in[7];
  const float* ln_z_w = (const float*)d_in[8];
  const float* ln_z_b = (const float*)d_in[9];
  float* out = (float*)d_out;

  m_ln_kernel<<<NN, 256, 0, stream>>>(m, ln_m_w, ln_m_b, out);
  // 2304 blocks * 8 waves = 18432 waves; 589824 rows -> exactly 32 rows/wave
  z_kernel<<<2304, 256, 0, stream>>>(z, x, lin_w, lin_b, ln_z_w, ln_z_b, out);
}